// PatchMatch_11931419149025
// MI455X (gfx1250) — compile-verified
//
#include <hip/hip_runtime.h>

// Problem constants (match reference)
#define BB 4
#define ND 16
#define HH 192
#define WW 256
#define NB 9
#define HW (HH * WW)  // 49152

__device__ __forceinline__ float pm_sigmoid(float d) {
  // d = |x1-x|*40, clipped to [0,4]; out = sigmoid(4-2d) = 1/(1+e^{2d-4})
  d = fminf(d, 4.0f);
  float e = __expf(2.0f * d - 4.0f);        // v_exp_f32 path
  return __fdividef(1.0f, 1.0f + e);        // v_rcp_f32 path
}

// ---------------------------------------------------------------------------
// Kernel 1: xT[b][h][w][n] = (1/depth - 1/dmax) / (1/dmin - 1/dmax)
// Channel-transposed layout so each bilinear corner is ONE 64B line.
// Reads coalesced (w across lanes), writes 64B/lane -> wave writes 2KB blocks.
// ---------------------------------------------------------------------------
__global__ void build_xt_kernel(const float* __restrict__ ds,
                                const float* __restrict__ dmin,
                                const float* __restrict__ dmax,
                                float* __restrict__ xt) {
  const int bh = blockIdx.x;
  const int b  = bh / HH;
  const int h  = bh - b * HH;
  const int w  = threadIdx.x;

  const float inv_min = 1.0f / dmin[b];
  const float inv_max = 1.0f / dmax[b];
  const float sc      = 1.0f / (inv_min - inv_max);

  const size_t sp = (size_t)h * WW + w;
  float v[ND];
#pragma unroll
  for (int n = 0; n < ND; ++n) {
    float d = __builtin_nontemporal_load(ds + (size_t)(b * ND + n) * HW + sp);
    v[n] = (1.0f / d - inv_max) * sc;
  }
  float4* o = reinterpret_cast<float4*>(xt + ((size_t)bh * WW + w) * ND);
#pragma unroll
  for (int g = 0; g < 4; ++g)
    o[g] = make_float4(v[4 * g + 0], v[4 * g + 1], v[4 * g + 2], v[4 * g + 3]);
}

// ---------------------------------------------------------------------------
// Kernel 2: per (b,h) row, lane = pixel w. Async-stage 9 grid slices into LDS,
// consume with progressive s_wait_asynccnt; gather 4 corners x 16ch (64B lines)
// from L2-resident xT; NT-store 16x9 outputs.
// ---------------------------------------------------------------------------
template <bool USE_XT>
__global__ void pm_eval_kernel(const float* __restrict__ xt,
                               const float* __restrict__ ds,
                               const float* __restrict__ grid,
                               const float* __restrict__ dmin,
                               const float* __restrict__ dmax,
                               float* __restrict__ out) {
  __shared__ float gsl[NB][2 * WW];  // 9 slices x 256 lanes x float2 = 18 KB

  const int bh = blockIdx.x;
  const int b  = bh / HH;
  const int h  = bh - b * HH;
  const int w  = threadIdx.x;

  // ---- issue 9 async grid fetches (CDNA5 GLOBAL_LOAD_ASYNC_TO_LDS) ----
  // grid element (b, k*HH + h, w, :) -> float2 per lane, contiguous per wave
  const float* gb = grid + (((size_t)b * NB * HH + h) * WW + w) * 2;
#pragma unroll
  for (int k = 0; k < NB; ++k) {
    unsigned int la =
        (unsigned int)(unsigned long long)(const void*)&gsl[k][2 * w];
    const void* ga = (const void*)(gb + (size_t)k * HW * 2);
    asm volatile("global_load_async_to_lds_b64 %0, %1, off"
                 :: "v"(la), "v"(ga)
                 : "memory");
  }

  // ---- center values (reused across all 9 neighbors) ----
  float ctr[ND];
  float inv_max = 0.0f, sc = 0.0f;
  if constexpr (USE_XT) {
    const float4* cp =
        reinterpret_cast<const float4*>(xt + ((size_t)bh * WW + w) * ND);
#pragma unroll
    for (int g = 0; g < 4; ++g) {
      float4 t = cp[g];
      ctr[4 * g + 0] = t.x; ctr[4 * g + 1] = t.y;
      ctr[4 * g + 2] = t.z; ctr[4 * g + 3] = t.w;
    }
  } else {
    const float inv_min = 1.0f / dmin[b];
    inv_max = 1.0f / dmax[b];
    sc      = 1.0f / (inv_min - inv_max);
    const size_t sp = (size_t)h * WW + w;
#pragma unroll
    for (int n = 0; n < ND; ++n) {
      float d = ds[(size_t)(b * ND + n) * HW + sp];
      ctr[n] = (1.0f / d - inv_max) * sc;
    }
  }

  auto body = [&](int k) {
    float2 g2 = *reinterpret_cast<const float2*>(&gsl[k][2 * w]);
    // align_corners=False unnormalize: ((x+1)*W - 1)/2, border clamp
    float ix = (g2.x + 1.0f) * (0.5f * WW) - 0.5f;
    float iy = (g2.y + 1.0f) * (0.5f * HH) - 0.5f;
    ix = fminf(fmaxf(ix, 0.0f), (float)(WW - 1));
    iy = fminf(fmaxf(iy, 0.0f), (float)(HH - 1));
    float fx0 = floorf(ix), fy0 = floorf(iy);
    float wx = ix - fx0, wy = iy - fy0;
    int ix0 = (int)fx0, iy0 = (int)fy0;
    int ix1 = min(ix0 + 1, WW - 1), iy1 = min(iy0 + 1, HH - 1);
    float w00 = (1.0f - wx) * (1.0f - wy), w01 = wx * (1.0f - wy);
    float w10 = (1.0f - wx) * wy,          w11 = wx * wy;

    // out index (b,n,k,h,w): n stride = NB*HW
    float* ob = out + (((size_t)b * ND * NB + k) * HH + h) * (size_t)WW + w;

    if constexpr (USE_XT) {
      const size_t rb = (size_t)b * HH;
      const float* p00 = xt + ((rb + iy0) * WW + ix0) * ND;
      const float* p01 = xt + ((rb + iy0) * WW + ix1) * ND;
      const float* p10 = xt + ((rb + iy1) * WW + ix0) * ND;
      const float* p11 = xt + ((rb + iy1) * WW + ix1) * ND;
#pragma unroll
      for (int g = 0; g < 4; ++g) {
        float4 v00 = *reinterpret_cast<const float4*>(p00 + 4 * g);
        float4 v01 = *reinterpret_cast<const float4*>(p01 + 4 * g);
        float4 v10 = *reinterpret_cast<const float4*>(p10 + 4 * g);
        float4 v11 = *reinterpret_cast<const float4*>(p11 + 4 * g);
        float i0 = w00 * v00.x + w01 * v01.x + w10 * v10.x + w11 * v11.x;
        float i1 = w00 * v00.y + w01 * v01.y + w10 * v10.y + w11 * v11.y;
        float i2 = w00 * v00.z + w01 * v01.z + w10 * v10.z + w11 * v11.z;
        float i3 = w00 * v00.w + w01 * v01.w + w10 * v10.w + w11 * v11.w;
        __builtin_nontemporal_store(pm_sigmoid(fabsf(i0 - ctr[4 * g + 0]) * 40.0f),
                                    ob + (size_t)(4 * g + 0) * NB * HW);
        __builtin_nontemporal_store(pm_sigmoid(fabsf(i1 - ctr[4 * g + 1]) * 40.0f),
                                    ob + (size_t)(4 * g + 1) * NB * HW);
        __builtin_nontemporal_store(pm_sigmoid(fabsf(i2 - ctr[4 * g + 2]) * 40.0f),
                                    ob + (size_t)(4 * g + 2) * NB * HW);
        __builtin_nontemporal_store(pm_sigmoid(fabsf(i3 - ctr[4 * g + 3]) * 40.0f),
                                    ob + (size_t)(4 * g + 3) * NB * HW);
      }
    } else {
#pragma unroll 4
      for (int n = 0; n < ND; ++n) {
        const float* ch = ds + (size_t)(b * ND + n) * HW;
        float v00 = (__fdividef(1.0f, ch[iy0 * WW + ix0]) - inv_max) * sc;
        float v01 = (__fdividef(1.0f, ch[iy0 * WW + ix1]) - inv_max) * sc;
        float v10 = (__fdividef(1.0f, ch[iy1 * WW + ix0]) - inv_max) * sc;
        float v11 = (__fdividef(1.0f, ch[iy1 * WW + ix1]) - inv_max) * sc;
        float itp = w00 * v00 + w01 * v01 + w10 * v10 + w11 * v11;
        __builtin_nontemporal_store(pm_sigmoid(fabsf(itp - ctr[n]) * 40.0f),
                                    ob + (size_t)n * NB * HW);
      }
    }
  };

  // Progressive waits: async loads complete in order, so before consuming
  // slice k we only need ASYNCcnt <= 8-k -> fetch of later slices overlaps
  // with gather/compute of earlier ones.
#define PM_STEP(k, wc)                                         \
  do {                                                         \
    asm volatile("s_wait_asynccnt " #wc ::: "memory");         \
    body(k);                                                   \
  } while (0)

  PM_STEP(0, 8); PM_STEP(1, 7); PM_STEP(2, 6);
  PM_STEP(3, 5); PM_STEP(4, 4); PM_STEP(5, 3);
  PM_STEP(6, 2); PM_STEP(7, 1); PM_STEP(8, 0);
#undef PM_STEP
}

// ---------------------------------------------------------------------------
extern "C" void kernel_launch(void* const* d_in, const int* in_sizes, int n_in,
                              void* d_out, int out_size, void* d_ws,
                              size_t ws_size, hipStream_t stream) {
  const float* ds   = (const float*)d_in[0];  // depth_sample (B,N,H,W)
  const float* dmin = (const float*)d_in[1];  // (B,)
  const float* dmax = (const float*)d_in[2];  // (B,)
  const float* grid = (const float*)d_in[3];  // (B,9H,W,2)
  float* out = (float*)d_out;                 // (B,N,9,H,W) f32

  const size_t need = (size_t)BB * HH * WW * ND * sizeof(float);  // 12 MB
  if (ws_size >= need) {
    float* xt = (float*)d_ws;  // (B,H,W,N) channel-transposed, L2-resident
    build_xt_kernel<<<BB * HH, WW, 0, stream>>>(ds, dmin, dmax, xt);
    pm_eval_kernel<true><<<BB * HH, WW, 0, stream>>>(xt, ds, grid, dmin, dmax, out);
  } else {
    pm_eval_kernel<false><<<BB * HH, WW, 0, stream>>>(nullptr, ds, grid, dmin, dmax, out);
  }
}